// WorkingFractionalPINO_48661979464035
// MI455X (gfx1250) — compile-verified
//
#include <hip/hip_runtime.h>
#include <math.h>

// ---------------------------------------------------------------------------
// FractionalPINO on MI455X (gfx1250, wave32).
// Bandwidth-bound: ~570MB of fp32 MLP weights read once -> ~25us floor at
// 23.3 TB/s.  GEMVs use V_WMMA_F32_16X16X4_F32 (the fp32 WMMA shape) with
// TWO independent accumulators to break the WMMA D->C serial chain
// (hazard table: f32 WMMA->WMMA RAW costs NOP slots), split-K +
// global_atomic_add_f32 for parallelism on skinny (2048-row) layers.
// ---------------------------------------------------------------------------

typedef float v2f __attribute__((ext_vector_type(2)));
typedef float v8f __attribute__((ext_vector_type(8)));

#define HW 128
#define HWHW 16384        // 128*128
#define MODES 16384
#define WIDTH 2048

// ---------------------------------------------------------------------------
// 3x3 SAME conv, NCHW, batch 1.  One thread per output element.
// ---------------------------------------------------------------------------
__global__ void conv3x3_kernel(const float* __restrict__ in,
                               const float* __restrict__ w,
                               const float* __restrict__ b,
                               float* __restrict__ out,
                               int Cin, int Cout) {
  int idx = blockIdx.x * blockDim.x + threadIdx.x;
  int total = Cout * HWHW;
  if (idx >= total) return;
  int co = idx / HWHW;
  int p  = idx - co * HWHW;
  int y  = p / HW;
  int x  = p - y * HW;
  float acc = b[co];
  for (int ci = 0; ci < Cin; ++ci) {
    const float* ip = in + ci * HWHW;
    const float* wp = w + (co * Cin + ci) * 9;
#pragma unroll
    for (int dy = -1; dy <= 1; ++dy) {
#pragma unroll
      for (int dx = -1; dx <= 1; ++dx) {
        int yy = y + dy, xx = x + dx;
        if (yy >= 0 && yy < HW && xx >= 0 && xx < HW)
          acc += ip[yy * HW + xx] * wp[(dy + 1) * 3 + (dx + 1)];
      }
    }
  }
  out[idx] = acc;
}

// ---------------------------------------------------------------------------
// BatchNorm statistics: one block (256 thr) per channel over 16384 elements.
// stats[2c] = mean, stats[2c+1] = rsqrt(var + eps)
// ---------------------------------------------------------------------------
__global__ void bn_stats_kernel(const float* __restrict__ y,
                                float* __restrict__ stats) {
  __shared__ float ssum[256];
  __shared__ float ssq[256];
  int c = blockIdx.x;
  float s = 0.f, q = 0.f;
  for (int i = threadIdx.x; i < HWHW; i += 256) {
    float v = y[c * HWHW + i];
    s += v; q += v * v;
  }
  ssum[threadIdx.x] = s; ssq[threadIdx.x] = q;
  __syncthreads();
  for (int st = 128; st > 0; st >>= 1) {
    if (threadIdx.x < st) {
      ssum[threadIdx.x] += ssum[threadIdx.x + st];
      ssq[threadIdx.x]  += ssq[threadIdx.x + st];
    }
    __syncthreads();
  }
  if (threadIdx.x == 0) {
    float m = ssum[0] * (1.0f / HWHW);
    float v = ssq[0] * (1.0f / HWHW) - m * m;
    stats[2 * c]     = m;
    stats[2 * c + 1] = rsqrtf(v + 1e-5f);
  }
}

__global__ void bn_apply_relu_kernel(float* __restrict__ y,
                                     const float* __restrict__ stats,
                                     const float* __restrict__ g,
                                     const float* __restrict__ b,
                                     int total) {
  int idx = blockIdx.x * blockDim.x + threadIdx.x;
  if (idx >= total) return;
  int c = idx / HWHW;
  float m = stats[2 * c], rs = stats[2 * c + 1];
  float v = (y[idx] - m) * rs * g[c] + b[c];
  y[idx] = fmaxf(v, 0.f);
}

// ---------------------------------------------------------------------------
// y[m] = bias[m]  (pre-init for split-K atomic accumulation)
// ---------------------------------------------------------------------------
__global__ void init_bias_kernel(float* __restrict__ y,
                                 const float* __restrict__ bias, int n) {
  int i = blockIdx.x * blockDim.x + threadIdx.x;
  if (i < n) y[i] = bias[i];
}

__global__ void relu_kernel(float* __restrict__ y, int n) {
  int i = blockIdx.x * blockDim.x + threadIdx.x;
  if (i < n) y[i] = fmaxf(y[i], 0.f);
}

// ---------------------------------------------------------------------------
// GEMV via V_WMMA_F32_16X16X4_F32.
//   Each wave owns 16 output rows; block (8 waves) owns 128 rows.
//   blockIdx.y selects a K-slice of length klen (split-K); partial results
//   are accumulated into y with global_atomic_add_f32.
// A-tile layout (32-bit A 16x4): lanes 0-15 -> M=lane, K=0..1 in v[0..1];
//                                lanes 16-31 -> M=lane-16, K=2..3.
// B-tile: input vector broadcast across all 16 N columns -> D column 0
//         (lanes 0 and 16) holds W.x for the 16 rows.
// Two accumulators on interleaved K-steps break the WMMA D->C RAW chain.
// ---------------------------------------------------------------------------
__global__ void __launch_bounds__(256)
gemv_wmma_kernel(const float* __restrict__ W,
                 const float* __restrict__ x,
                 float* __restrict__ y,
                 int in_d, int klen) {
  const int lane = threadIdx.x & 31;
  const int wave = threadIdx.x >> 5;
  const int row0 = (blockIdx.x * 8 + wave) * 16;
  const int mrow = row0 + (lane & 15);
  const int khalf = (lane >> 4) << 1;                 // 0 or 2
  const int k0 = blockIdx.y * klen;

  const float* wrow = W + (size_t)mrow * (size_t)in_d + khalf;
  const float* xoff = x + khalf;

  v8f acc0 = {};
  v8f acc1 = {};
  for (int kk = k0; kk < k0 + klen; kk += 64) {
    // stay ~2KB ahead of the weight stream
    __builtin_prefetch((const void*)(wrow + kk + 512), 0, 1);
#pragma unroll
    for (int k = kk; k < kk + 64; k += 8) {
      v2f a0 = *(const v2f*)(wrow + k);
      v2f b0 = *(const v2f*)(xoff + k);
      v2f a1 = *(const v2f*)(wrow + k + 4);
      v2f b1 = *(const v2f*)(xoff + k + 4);
      acc0 = __builtin_amdgcn_wmma_f32_16x16x4_f32(
          false, a0, false, b0, (short)0, acc0, false, false);
      acc1 = __builtin_amdgcn_wmma_f32_16x16x4_f32(
          false, a1, false, b1, (short)0, acc1, false, false);
    }
  }

  // Column N=0 of D: lane 0 -> rows row0..row0+7, lane 16 -> rows row0+8..+15
  if (lane == 0 || lane == 16) {
    int base = row0 + ((lane >> 4) << 3);
#pragma unroll
    for (int r = 0; r < 8; ++r)
      atomicAdd(&y[base + r], acc0[r] + acc1[r]);
  }
}

// ---------------------------------------------------------------------------
// Grunwald-Letnikov weights, closed form (fully parallel, alpha = 0.5):
//   w[j] = Gamma(j-a) / (Gamma(-a) * j!),  sign negative for j>=1.
//   |Gamma(-0.5)| = 2*sqrt(pi) -> log = 1.2655121
// ---------------------------------------------------------------------------
__global__ void gl_weights_kernel(float* __restrict__ w, int n) {
  int j = blockIdx.x * blockDim.x + threadIdx.x;
  if (j >= n) return;
  if (j == 0) { w[0] = 1.0f; return; }
  const float alpha = 0.5f;
  float lg = lgammaf((float)j - alpha) - lgammaf((float)j + 1.0f) - 1.2655121f;
  w[j] = -expf(lg);
}

// ---------------------------------------------------------------------------
// Direct causal convolution == linear conv via FFT in the reference.
//   y[i] = scale * sum_{j<=i} glw[j] * x[i-j]
// x and glw are 64KB each -> fully L2 resident.
// ---------------------------------------------------------------------------
__global__ void frac_conv_kernel(const float* __restrict__ x,
                                 const float* __restrict__ glw,
                                 float* __restrict__ y,
                                 int n, float scale) {
  int i = blockIdx.x * blockDim.x + threadIdx.x;
  if (i >= n) return;
  float acc = 0.f;
  for (int j = 0; j <= i; ++j)
    acc = fmaf(glw[j], x[i - j], acc);
  y[i] = acc * scale;
}

// ---------------------------------------------------------------------------
// Host-side orchestration
// ---------------------------------------------------------------------------
static inline void launch_gemv(const float* W, const float* x,
                               const float* bias, float* y,
                               int out_d, int in_d, int splitk, int do_relu,
                               hipStream_t stream) {
  init_bias_kernel<<<(out_d + 255) / 256, 256, 0, stream>>>(y, bias, out_d);
  dim3 grid(out_d / 128, splitk);
  gemv_wmma_kernel<<<grid, 256, 0, stream>>>(W, x, y, in_d, in_d / splitk);
  if (do_relu)
    relu_kernel<<<(out_d + 255) / 256, 256, 0, stream>>>(y, out_d);
}

extern "C" void kernel_launch(void* const* d_in, const int* in_sizes, int n_in,
                              void* d_out, int out_size, void* d_ws, size_t ws_size,
                              hipStream_t stream) {
  const float* x       = (const float*)d_in[0];
  const float* conv1_w = (const float*)d_in[1];
  const float* conv1_b = (const float*)d_in[2];
  const float* bn1_g   = (const float*)d_in[3];
  const float* bn1_b   = (const float*)d_in[4];
  const float* conv2_w = (const float*)d_in[5];
  const float* conv2_b = (const float*)d_in[6];
  const float* bn2_g   = (const float*)d_in[7];
  const float* bn2_b   = (const float*)d_in[8];
  const float* conv3_w = (const float*)d_in[9];
  const float* conv3_b = (const float*)d_in[10];
  const float* enc_w1  = (const float*)d_in[11];
  const float* enc_b1  = (const float*)d_in[12];
  const float* enc_w2  = (const float*)d_in[13];
  const float* enc_b2  = (const float*)d_in[14];
  const float* enc_w3  = (const float*)d_in[15];
  const float* enc_b3  = (const float*)d_in[16];
  const float* op_w1   = (const float*)d_in[17];
  const float* op_b1   = (const float*)d_in[18];
  const float* op_w2   = (const float*)d_in[19];
  const float* op_b2   = (const float*)d_in[20];
  const float* op_w3   = (const float*)d_in[21];
  const float* op_b3   = (const float*)d_in[22];

  float* ws = (float*)d_ws;
  float* y1    = ws;                 // 16*16384 = 262144
  float* y2    = y1 + 262144;        // 32*16384 = 524288
  float* xf    = y2 + 524288;        // 16384
  float* stats = xf + 16384;         // 64
  float* glw   = stats + 64;         // 16384
  float* h1    = glw + 16384;        // 2048
  float* h2    = h1 + 2048;          // 2048
  float* v1    = h2 + 2048;          // 16384
  float* vp    = v1 + 16384;         // 16384
  float* g1    = vp + 16384;         // 2048
  float* g2    = g1 + 2048;          // 2048
  float* out   = (float*)d_out;      // 16384

  // ---- conv stack --------------------------------------------------------
  conv3x3_kernel<<<(16 * HWHW + 255) / 256, 256, 0, stream>>>(x, conv1_w, conv1_b, y1, 1, 16);
  bn_stats_kernel<<<16, 256, 0, stream>>>(y1, stats);
  bn_apply_relu_kernel<<<(16 * HWHW + 255) / 256, 256, 0, stream>>>(y1, stats, bn1_g, bn1_b, 16 * HWHW);

  conv3x3_kernel<<<(32 * HWHW + 255) / 256, 256, 0, stream>>>(y1, conv2_w, conv2_b, y2, 16, 32);
  bn_stats_kernel<<<32, 256, 0, stream>>>(y2, stats);
  bn_apply_relu_kernel<<<(32 * HWHW + 255) / 256, 256, 0, stream>>>(y2, stats, bn2_g, bn2_b, 32 * HWHW);

  conv3x3_kernel<<<(HWHW + 255) / 256, 256, 0, stream>>>(y2, conv3_w, conv3_b, xf, 32, 1);

  // ---- encoder MLP (WMMA GEMVs) -----------------------------------------
  launch_gemv(enc_w1, xf, enc_b1, h1, WIDTH, MODES, /*splitk=*/8, /*relu=*/1, stream);
  launch_gemv(enc_w2, h1, enc_b2, h2, WIDTH, WIDTH, /*splitk=*/8, /*relu=*/1, stream);
  launch_gemv(enc_w3, h2, enc_b3, v1, MODES, WIDTH, /*splitk=*/1, /*relu=*/0, stream);

  // ---- fractional derivative --------------------------------------------
  gl_weights_kernel<<<(MODES + 255) / 256, 256, 0, stream>>>(glw, MODES);
  const float scale = powf((float)(MODES - 1), 0.5f);   // h^{-alpha}, alpha=0.5
  frac_conv_kernel<<<(MODES + 255) / 256, 256, 0, stream>>>(v1, glw, vp, MODES, scale);

  // ---- operator MLP (WMMA GEMVs) ----------------------------------------
  launch_gemv(op_w1, vp, op_b1, g1, WIDTH, MODES, /*splitk=*/8, /*relu=*/1, stream);
  launch_gemv(op_w2, g1, op_b2, g2, WIDTH, WIDTH, /*splitk=*/8, /*relu=*/1, stream);
  launch_gemv(op_w3, g2, op_b3, out, MODES, WIDTH, /*splitk=*/1, /*relu=*/0, stream);

  (void)in_sizes; (void)n_in; (void)out_size; (void)ws_size;
}